// NGramRepeatBlock_15650860826872
// MI455X (gfx1250) — compile-verified
//
#include <hip/hip_runtime.h>
#include <hip/hip_bf16.h>
#include <stdint.h>

// Native 128-bit float vector (HIP's float4 is a struct and is rejected by
// __builtin_nontemporal_load; ext_vector_type is a true vector type).
typedef float f32x4 __attribute__((ext_vector_type(4)));

// ---------------------------------------------------------------------------
// CDNA5 async global->LDS support (gfx1250). Builtin signature (from hipcc
// diagnostics): (as1 int* gsrc, as3 int* lds_dst, imm offset, imm cpol).
// ---------------------------------------------------------------------------
#if defined(__has_builtin)
#if __has_builtin(__builtin_amdgcn_global_load_async_to_lds_b32) && \
    __has_builtin(__builtin_amdgcn_s_wait_asynccnt)
#define USE_ASYNC_LDS 1
#endif
#endif
#ifndef USE_ASYNC_LDS
#define USE_ASYNC_LDS 0
#endif

typedef __attribute__((address_space(1))) int gint_t;
typedef __attribute__((address_space(3))) int lint_t;

__device__ __forceinline__ void lds_copy_b32(const int* gsrc, int* lds_dst) {
#if USE_ASYNC_LDS
    __builtin_amdgcn_global_load_async_to_lds_b32(
        (gint_t*)(gsrc), (lint_t*)(lds_dst), 0, 0);
#else
    *lds_dst = *gsrc;
#endif
}

__device__ __forceinline__ void lds_copy_wait() {
#if USE_ASYNC_LDS
    __builtin_amdgcn_s_wait_asynccnt(0);
#endif
    __syncthreads();
}

// ---------------------------------------------------------------------------
// Kernel 1: streaming copy lprobs -> d_out (bandwidth-dominant part).
// B128 vectorized; nontemporal loads (read-once stream) so we do not evict
// useful L2 lines; regular-temporal stores so the subsequent scattered -inf
// writes from the ban kernel hit L2.
// ---------------------------------------------------------------------------
__global__ __launch_bounds__(256)
void copy_f32_kernel(const f32x4* __restrict__ in, f32x4* __restrict__ out,
                     long n4, const float* __restrict__ in_tail,
                     float* __restrict__ out_tail, int rem) {
    long i = (long)blockIdx.x * blockDim.x + threadIdx.x;
    if (i < n4) {
        // speculative prefetch ahead (lowered to global_prefetch_b8)
        if (i + 4096 < n4)
            __builtin_prefetch((const void*)(in + i + 4096), 0, 0);
        f32x4 v = __builtin_nontemporal_load(in + i);
        out[i] = v;
    }
    if (i == 0) {
        for (int k = 0; k < rem; ++k) out_tail[k] = in_tail[k];
    }
}

// ---------------------------------------------------------------------------
// Kernel 2: n-gram repeat blocking. One workgroup per row (grid-stride over
// rows since B is only known on-device). Token row is staged into LDS via
// CDNA5 async global->LDS in tiles; every thread then scans windows and
// stamps -inf on matches.
// ---------------------------------------------------------------------------
#define NG_TILE 2048
#define NG_MAXM 15

__global__ __launch_bounds__(256)
void ngram_ban_kernel(const int* __restrict__ tokens, float* __restrict__ out,
                      const int* __restrict__ bsz_p, const int* __restrict__ step_p,
                      const int* __restrict__ beam_p, const int* __restrict__ n_p,
                      long tokens_total, long out_total) {
    __shared__ int s_tok[NG_TILE + NG_MAXM + 1];

    const int bsz  = *bsz_p;
    const int step = *step_p;
    const int beam = *beam_p;
    const int n    = *n_p;
    const int B = bsz * beam;
    if (B <= 0) return;
    const long SEQ = tokens_total / B;
    const long V   = out_total / B;
    const int m = n - 1;
    if (n <= 0 || step + 1 < n || m > NG_MAXM) return;

    const int num_win = step + 2 - n;      // windows j in [0, num_win)
    const int bdim = blockDim.x;
    const int tid  = threadIdx.x;

    for (int row = blockIdx.x; row < B; row += gridDim.x) {
        const int* trow = tokens + (long)row * SEQ;
        float*     orow = out    + (long)row * V;

        // current (n-1)-gram suffix: tokens[step-m+1 .. step]
        int cur[NG_MAXM];
        for (int k = 0; k < m; ++k) cur[k] = trow[step - m + 1 + k];

        for (int base = 0; base < num_win; base += NG_TILE) {
            const int wcount = min(NG_TILE, num_win - base);
            const int lcount = wcount + m;     // tokens [base, base+wcount+m)

            // async stage tile of the token row into LDS
            for (int i = tid; i < lcount; i += bdim)
                lds_copy_b32(trow + base + i, &s_tok[i]);
            lds_copy_wait();

            for (int j = tid; j < wcount; j += bdim) {
                bool match = true;
#pragma unroll 4
                for (int k = 0; k < m; ++k)
                    match &= (s_tok[j + k] == cur[k]);
                if (match) {
                    int t = s_tok[j + m];      // historically-following token
                    if (t >= 0 && (long)t < V)
                        orow[t] = -__builtin_huge_valf();
                }
            }
            __syncthreads();   // protect LDS before next tile
        }
    }
}

// ---------------------------------------------------------------------------
// Launcher. Inputs (setup_inputs order):
//   d_in[0] tokens (int32, B*SEQ), d_in[1] lprobs (f32, B*V),
//   d_in[2] bsz, d_in[3] step, d_in[4] beam_size, d_in[5] no_repeat_ngram_size
//   (scalars are 1-element device int arrays).
// ---------------------------------------------------------------------------
extern "C" void kernel_launch(void* const* d_in, const int* in_sizes, int n_in,
                              void* d_out, int out_size, void* d_ws, size_t ws_size,
                              hipStream_t stream) {
    const int*   tokens = (const int*)d_in[0];
    const float* lprobs = (const float*)d_in[1];
    const int*   bsz_p  = (const int*)d_in[2];
    const int*   step_p = (const int*)d_in[3];
    const int*   beam_p = (const int*)d_in[4];
    const int*   n_p    = (const int*)d_in[5];

    const long total = (long)out_size;
    const long n4    = total / 4;
    const int  rem   = (int)(total % 4);

    const int threads = 256;
    long blocks = (n4 + threads - 1) / threads;
    if (blocks < 1) blocks = 1;

    copy_f32_kernel<<<(unsigned)blocks, threads, 0, stream>>>(
        (const f32x4*)lprobs, (f32x4*)d_out, n4,
        lprobs + n4 * 4, (float*)d_out + n4 * 4, rem);

    const long tokens_total = (long)in_sizes[0];
    ngram_ban_kernel<<<1024, 256, 0, stream>>>(
        tokens, (float*)d_out, bsz_p, step_p, beam_p, n_p,
        tokens_total, total);
}